// DyResGEN_36490042147128
// MI455X (gfx1250) — compile-verified
//
#include <hip/hip_runtime.h>
#include <math.h>

typedef float v2f __attribute__((ext_vector_type(2)));
typedef float v8f __attribute__((ext_vector_type(8)));

#define MSG_EPS 1e-7f

// ---------------------------------------------------------------------------
// WMMA f32 GEMM:  out[n x Nout] = A[n x K] @ W[K x Nout] (+bias)(+residual)(relu)
// Block = 8 waves, 128 rows x 32 cols (two 16-col tiles per wave).
// The B panel is staged in LDS in per-lane consumption order so each B operand
// is one 8-byte-aligned ds_load_b64 into an even VGPR pair (no marshalling
// movs, zero bank conflicts: lane l reads dwords {2l, 2l+1}).
// A loads are unconditional v2f (row-clamped; tail rows computed, not stored).
// ---------------------------------------------------------------------------
__global__ void k_gemm(const float* __restrict__ A, const float* __restrict__ W,
                       const float* __restrict__ bias, const float* __restrict__ R,
                       float* __restrict__ out, int n, int K, int Nout, int relu)
{
    __shared__ float sW[256 * 32];                       // K<=256 -> <=32KB
    const int tid = threadIdx.x;
    const int n0  = blockIdx.y * 32;
    // slot s: inner = s&1, lane = (s>>1)&31, T = (s>>6)&1, i = s>>7
    // kb = i*4 + (lane>>4)*2 + inner ; col = n0 + T*16 + (lane&15)
    for (int s = tid; s < K * 32; s += 256) {
        int inner = s & 1;
        int lane  = (s >> 1) & 31;
        int T     = (s >> 6) & 1;
        int i     = s >> 7;
        int kb    = i * 4 + ((lane >> 4) << 1) + inner;
        int col   = n0 + T * 16 + (lane & 15);
        sW[s] = W[(size_t)kb * Nout + col];
    }
    __syncthreads();

    const int wave = tid >> 5, lane = tid & 31;
    const int m = lane & 15, half = lane >> 4;           // ISA 16x4 f32 A layout
    const int rowBase = blockIdx.x * 128 + wave * 16;
    int row = rowBase + m;
    if (row >= n) row = n - 1;                           // clamp: loads unconditional
    const float* Ar = A + (size_t)row * K + half * 2;
    const float* sB = sW + lane * 2;

    v8f acc0 = {0.f, 0.f, 0.f, 0.f, 0.f, 0.f, 0.f, 0.f};
    v8f acc1 = {0.f, 0.f, 0.f, 0.f, 0.f, 0.f, 0.f, 0.f};
    const int iters = K >> 2;
    for (int i = 0; i < iters; ++i) {
        v2f a  = *(const v2f*)(Ar + i * 4);              // global_load_b64
        v2f b0 = *(const v2f*)(sB + (i * 2 + 0) * 64);   // ds_load_b64
        v2f b1 = *(const v2f*)(sB + (i * 2 + 1) * 64);   // ds_load_b64
        acc0 = __builtin_amdgcn_wmma_f32_16x16x4_f32(false, a, false, b0,
                                                     (short)0, acc0, false, false);
        acc1 = __builtin_amdgcn_wmma_f32_16x16x4_f32(false, a, false, b1,
                                                     (short)0, acc1, false, false);
    }

    const int col0 = n0 + m;
    const int col1 = n0 + 16 + m;
    #pragma unroll
    for (int i = 0; i < 8; ++i) {                        // D: VGPR i -> M = i + half*8
        const int rr = rowBase + half * 8 + i;
        if (rr < n) {
            float v0 = acc0[i];
            float v1 = acc1[i];
            if (bias) { v0 += bias[col0]; v1 += bias[col1]; }
            if (R)    { v0 += R[(size_t)rr * Nout + col0];
                        v1 += R[(size_t)rr * Nout + col1]; }
            if (relu) { v0 = fmaxf(v0, 0.f); v1 = fmaxf(v1, 0.f); }
            out[(size_t)rr * Nout + col0] = v0;
            out[(size_t)rr * Nout + col1] = v1;
        }
    }
}

// ---------------------------------------------------------------------------
// Edge pass A: segment max of logits (all logits > 0 -> int-bit atomicMax valid)
// m = relu(h[src] + ea0*We0 + ea1*We1) + eps ; logit = t*m
// ---------------------------------------------------------------------------
__global__ void k_edge_max(const int* __restrict__ src, const int* __restrict__ dst,
                           const int* __restrict__ emask, const float* __restrict__ eattr,
                           const float* __restrict__ h, const float* __restrict__ Wedge,
                           const float* __restrict__ tptr, float* __restrict__ mx,
                           int E, int d)
{
    int e = blockIdx.x * 256 + threadIdx.x;
    if (e >= E) return;
    __builtin_prefetch(&src[(e + 4096 < E) ? e + 4096 : e], 0, 0);
    if (!emask[e]) return;
    const int j = blockIdx.y * 4;
    const float t = *tptr;
    const int s = src[e], dn = dst[e];
    const float ea0 = eattr[2 * e], ea1 = eattr[2 * e + 1];
    const float4 hv = *(const float4*)(h + (size_t)s * d + j);
    const float4 w0 = *(const float4*)(Wedge + j);
    const float4 w1 = *(const float4*)(Wedge + d + j);
    float m0 = fmaxf(hv.x + ea0 * w0.x + ea1 * w1.x, 0.f) + MSG_EPS;
    float m1 = fmaxf(hv.y + ea0 * w0.y + ea1 * w1.y, 0.f) + MSG_EPS;
    float m2 = fmaxf(hv.z + ea0 * w0.z + ea1 * w1.z, 0.f) + MSG_EPS;
    float m3 = fmaxf(hv.w + ea0 * w0.w + ea1 * w1.w, 0.f) + MSG_EPS;
    int* mp = (int*)(mx + (size_t)dn * d + j);
    atomicMax(&mp[0], __float_as_int(t * m0));
    atomicMax(&mp[1], __float_as_int(t * m1));
    atomicMax(&mp[2], __float_as_int(t * m2));
    atomicMax(&mp[3], __float_as_int(t * m3));
}

// ---------------------------------------------------------------------------
// Edge pass B: den += exp(logit - mx[dst]); num += m * exp(...)
// ---------------------------------------------------------------------------
__global__ void k_edge_sum(const int* __restrict__ src, const int* __restrict__ dst,
                           const int* __restrict__ emask, const float* __restrict__ eattr,
                           const float* __restrict__ h, const float* __restrict__ Wedge,
                           const float* __restrict__ tptr, const float* __restrict__ mx,
                           float* __restrict__ den, float* __restrict__ num, int E, int d)
{
    int e = blockIdx.x * 256 + threadIdx.x;
    if (e >= E || !emask[e]) return;
    const int j = blockIdx.y * 4;
    const float t = *tptr;
    const int s = src[e], dn = dst[e];
    const float ea0 = eattr[2 * e], ea1 = eattr[2 * e + 1];
    const float4 hv  = *(const float4*)(h + (size_t)s * d + j);
    const float4 w0  = *(const float4*)(Wedge + j);
    const float4 w1  = *(const float4*)(Wedge + d + j);
    const float4 mxv = *(const float4*)(mx + (size_t)dn * d + j);
    float m0 = fmaxf(hv.x + ea0 * w0.x + ea1 * w1.x, 0.f) + MSG_EPS;
    float m1 = fmaxf(hv.y + ea0 * w0.y + ea1 * w1.y, 0.f) + MSG_EPS;
    float m2 = fmaxf(hv.z + ea0 * w0.z + ea1 * w1.z, 0.f) + MSG_EPS;
    float m3 = fmaxf(hv.w + ea0 * w0.w + ea1 * w1.w, 0.f) + MSG_EPS;
    float e0 = __expf(t * m0 - mxv.x);
    float e1 = __expf(t * m1 - mxv.y);
    float e2 = __expf(t * m2 - mxv.z);
    float e3 = __expf(t * m3 - mxv.w);
    float* dp = den + (size_t)dn * d + j;
    float* np = num + (size_t)dn * d + j;
    __hip_atomic_fetch_add(&dp[0], e0, __ATOMIC_RELAXED, __HIP_MEMORY_SCOPE_AGENT);
    __hip_atomic_fetch_add(&dp[1], e1, __ATOMIC_RELAXED, __HIP_MEMORY_SCOPE_AGENT);
    __hip_atomic_fetch_add(&dp[2], e2, __ATOMIC_RELAXED, __HIP_MEMORY_SCOPE_AGENT);
    __hip_atomic_fetch_add(&dp[3], e3, __ATOMIC_RELAXED, __HIP_MEMORY_SCOPE_AGENT);
    __hip_atomic_fetch_add(&np[0], m0 * e0, __ATOMIC_RELAXED, __HIP_MEMORY_SCOPE_AGENT);
    __hip_atomic_fetch_add(&np[1], m1 * e1, __ATOMIC_RELAXED, __HIP_MEMORY_SCOPE_AGENT);
    __hip_atomic_fetch_add(&np[2], m2 * e2, __ATOMIC_RELAXED, __HIP_MEMORY_SCOPE_AGENT);
    __hip_atomic_fetch_add(&np[3], m3 * e3, __ATOMIC_RELAXED, __HIP_MEMORY_SCOPE_AGENT);
}

// o = h + num/(den + 1e-16)
__global__ void k_node_o(const float* __restrict__ h, const float* __restrict__ num,
                         const float* __restrict__ den, float* __restrict__ o, int cnt)
{
    int i = blockIdx.x * 256 + threadIdx.x;
    if (i < cnt) o[i] = h[i] + num[i] / (den[i] + 1e-16f);
}

__global__ void k_fill_f(float* p, float v, int cnt)
{
    int i = blockIdx.x * 256 + threadIdx.x;
    if (i < cnt) p[i] = v;
}
__global__ void k_fill_i(int* p, int v, int cnt)
{
    int i = blockIdx.x * 256 + threadIdx.x;
    if (i < cnt) p[i] = v;
}

__global__ void k_copy_edges(const int* __restrict__ eidx, int* src, int* dst,
                             int* emask, int E)
{
    int e = blockIdx.x * 256 + threadIdx.x;
    if (e >= E) return;
    src[e] = eidx[e];
    dst[e] = eidx[E + e];
    emask[e] = 1;
}

// LayerNorm (+optional ReLU), one wave32 per row
__global__ void k_layernorm(const float* __restrict__ x, float* __restrict__ y,
                            const float* __restrict__ g, const float* __restrict__ b,
                            int n, int d, int relu)
{
    int wave = threadIdx.x >> 5, lane = threadIdx.x & 31;
    int row = blockIdx.x * 8 + wave;
    if (row >= n) return;
    const float* xr = x + (size_t)row * d;
    float s = 0.f;
    for (int j = lane; j < d; j += 32) s += xr[j];
    for (int o = 16; o > 0; o >>= 1) s += __shfl_xor(s, o, 32);
    float mu = s / d;
    float v = 0.f;
    for (int j = lane; j < d; j += 32) { float t = xr[j] - mu; v += t * t; }
    for (int o = 16; o > 0; o >>= 1) v += __shfl_xor(v, o, 32);
    float inv = rsqrtf(v / d + 1e-5f);
    float* yr = y + (size_t)row * d;
    for (int j = lane; j < d; j += 32) {
        float o2 = (xr[j] - mu) * inv * g[j] + b[j];
        yr[j] = relu ? fmaxf(o2, 0.f) : o2;
    }
}

// invnorm = 1/||w||
__global__ void k_wnorm(const float* __restrict__ w, int d, float* ctrlF)
{
    __shared__ float ss[128];
    int t = threadIdx.x;
    ss[t] = (t < d) ? w[t] * w[t] : 0.f;
    __syncthreads();
    for (int o = 64; o > 0; o >>= 1) { if (t < o) ss[t] += ss[t + o]; __syncthreads(); }
    if (t == 0) ctrlF[8] = rsqrtf(ss[0]);
}

// score = tanh(x.w * invnorm); key = order-monotonic uint of score
__global__ void k_score(const float* __restrict__ x, const float* __restrict__ w,
                        const float* __restrict__ ctrlF, float* __restrict__ score,
                        unsigned* __restrict__ keys, int n, int d)
{
    int i = blockIdx.x * 256 + threadIdx.x;
    if (i >= n) return;
    const float* xr = x + (size_t)i * d;
    float dot = 0.f;
    for (int j = 0; j < d; ++j) dot += xr[j] * w[j];
    float s = tanhf(dot * ctrlF[8]);
    score[i] = s;
    unsigned u = __float_as_uint(s);
    keys[i] = (u & 0x80000000u) ? ~u : (u | 0x80000000u);
}

__global__ void k_topk_init(unsigned* ctrl, unsigned* hist)
{
    int t = threadIdx.x;
    if (t < 4) ctrl[t] = 0u;
    hist[t] = 0u;                                        // blockDim == 256
}

// byte-wise radix histogram over keys matching current prefix
__global__ void k_hist(const unsigned* __restrict__ keys, unsigned* __restrict__ hist,
                       const unsigned* __restrict__ ctrl, int n, int level)
{
    int i = blockIdx.x * 256 + threadIdx.x;
    if (i >= n) return;
    unsigned key = keys[i];
    if (level > 0) {
        unsigned mask = 0xFFFFFFFFu << (32 - 8 * level);
        if ((key & mask) != (ctrl[0] & mask)) return;
    }
    unsigned b = (key >> (24 - 8 * level)) & 255u;
    atomicAdd(&hist[b], 1u);
}

// pick the byte bin containing the k-th largest; accumulate threshold prefix
__global__ void k_hist_step(unsigned* hist, unsigned* ctrl, int k, int level)
{
    if (threadIdx.x == 0) {
        unsigned prefixAbove = ctrl[1];
        int kp = k - (int)prefixAbove;
        unsigned cum = 0, sab = 0;
        int bstar = 0;
        for (int b = 255; b >= 0; --b) {
            unsigned c = hist[b];
            if ((int)(cum + c) >= kp) { bstar = b; sab = cum; break; }
            cum += c;
        }
        ctrl[0] |= ((unsigned)bstar) << (24 - 8 * level);
        ctrl[1] = prefixAbove + sab;
        if (level == 3) ctrl[2] = (unsigned)(kp - (int)sab);   // tiesNeeded
    }
    __syncthreads();
    for (int i = threadIdx.x; i < 256; i += blockDim.x) hist[i] = 0u;
}

__global__ void k_sel_count(const unsigned* __restrict__ keys, const unsigned* __restrict__ ctrl,
                            unsigned* bcg, unsigned* bct, int n)
{
    __shared__ unsigned sg, st;
    if (threadIdx.x == 0) { sg = 0u; st = 0u; }
    __syncthreads();
    int i = blockIdx.x * 256 + threadIdx.x;
    if (i < n) {
        unsigned key = keys[i], thr = ctrl[0];
        if (key > thr) atomicAdd(&sg, 1u);
        else if (key == thr) atomicAdd(&st, 1u);
    }
    __syncthreads();
    if (threadIdx.x == 0) { bcg[blockIdx.x] = sg; bct[blockIdx.x] = st; }
}

__global__ void k_sel_scan(const unsigned* bcg, const unsigned* bct,
                           unsigned* bbg, unsigned* bbt, unsigned* ctrl, int nb)
{
    unsigned ag = 0, at = 0;
    for (int i = 0; i < nb; ++i) {
        bbg[i] = ag; ag += bcg[i];
        bbt[i] = at; at += bct[i];
    }
    ctrl[3] = ag;                                        // total "greater" count
}

__global__ void k_sel_write(const unsigned* __restrict__ keys, const float* __restrict__ score,
                            const unsigned* __restrict__ ctrl, const unsigned* __restrict__ bbg,
                            const unsigned* __restrict__ bbt, int* __restrict__ idxnew,
                            float* __restrict__ vals, int n)
{
    __shared__ unsigned char fg[256], ft[256];
    __shared__ unsigned pg[256], pt[256];
    int t = threadIdx.x;
    int i = blockIdx.x * 256 + t;
    unsigned thr = ctrl[0];
    int g = 0, ti = 0;
    if (i < n) { unsigned key = keys[i]; g = key > thr; ti = (key == thr); }
    fg[t] = (unsigned char)g; ft[t] = (unsigned char)ti;
    __syncthreads();
    if (t == 0) {
        unsigned a = 0, b = 0;
        for (int j = 0; j < 256; ++j) { pg[j] = a; a += fg[j]; pt[j] = b; b += ft[j]; }
    }
    __syncthreads();
    unsigned totG = ctrl[3], tneed = ctrl[2];
    if (g) {
        unsigned pos = bbg[blockIdx.x] + pg[t];
        idxnew[pos] = i; vals[pos] = score[i];
    } else if (ti) {
        unsigned gr = bbt[blockIdx.x] + pt[t];
        if (gr < tneed) {
            unsigned pos = totG + gr;
            idxnew[pos] = i; vals[pos] = score[i];
        }
    }
}

// gather kept rows, gate by tanh score, build inverse map
__global__ void k_gather_gate(const float* __restrict__ xin, float* __restrict__ xout,
                              const int* __restrict__ idxnew, const float* __restrict__ vals,
                              int* __restrict__ inv, int d)
{
    int i = blockIdx.x;
    int old = idxnew[i];
    float g = vals[i];
    for (int j = threadIdx.x; j < d; j += blockDim.x)
        xout[(size_t)i * d + j] = xin[(size_t)old * d + j] * g;
    if (threadIdx.x == 0) inv[old] = i;
}

__global__ void k_relabel(int* src, int* dst, int* emask, const int* __restrict__ inv, int E)
{
    int e = blockIdx.x * 256 + threadIdx.x;
    if (e >= E) return;
    int ns = inv[src[e]], nd = inv[dst[e]];
    int ok = emask[e] & (ns >= 0) & (nd >= 0);
    emask[e] = ok;
    src[e] = ns < 0 ? 0 : ns;
    dst[e] = nd < 0 ? 0 : nd;
}

// global mean pool (n x 128) then 128->96(relu)->64(relu)->2
__global__ void k_head(const float* __restrict__ x, int n,
                       const float* __restrict__ W2, const float* __restrict__ b2,
                       const float* __restrict__ W1, const float* __restrict__ b1,
                       const float* __restrict__ W0, const float* __restrict__ b0,
                       float* __restrict__ out)
{
    __shared__ float m[128], h2[96], h1[64];
    int t = threadIdx.x;
    if (t < 128) {
        float s = 0.f;
        for (int i = 0; i < n; ++i) s += x[(size_t)i * 128 + t];
        m[t] = s / n;
    }
    __syncthreads();
    if (t < 96) {
        float s = b2[t];
        for (int j = 0; j < 128; ++j) s += m[j] * W2[j * 96 + t];
        h2[t] = fmaxf(s, 0.f);
    }
    __syncthreads();
    if (t < 64) {
        float s = b1[t];
        for (int j = 0; j < 96; ++j) s += h2[j] * W1[j * 64 + t];
        h1[t] = fmaxf(s, 0.f);
    }
    __syncthreads();
    if (t < 2) {
        float s = b0[t];
        for (int j = 0; j < 64; ++j) s += h1[j] * W0[j * 2 + t];
        out[t] = s;
    }
}

// ---------------------------------------------------------------------------
extern "C" void kernel_launch(void* const* d_in, const int* in_sizes, int n_in,
                              void* d_out, int out_size, void* d_ws, size_t ws_size,
                              hipStream_t stream)
{
    (void)in_sizes; (void)n_in; (void)out_size; (void)ws_size;

    constexpr int  EE   = 800000;
    constexpr int  MAXN = 50000;
    constexpr int  MAXD = 128;
    constexpr size_t ND = (size_t)MAXN * MAXD;           // 6.4M floats per big buffer

    char* ws = (char*)d_ws;
    unsigned* ctrlU = (unsigned*)(ws + 0);
    float*    ctrlF = (float*)(ws + 0);
    unsigned* HIST  = (unsigned*)(ws + 4096);
    unsigned* BCG   = (unsigned*)(ws + 8192);
    unsigned* BCT   = (unsigned*)(ws + 9216);
    unsigned* BBG   = (unsigned*)(ws + 10240);
    unsigned* BBT   = (unsigned*)(ws + 11264);
    float*    SCOREb = (float*)(ws + 16384);
    unsigned* KEYSb  = (unsigned*)(ws + 16384 + 1 * 200000);
    float*    VALSb  = (float*)(ws + 16384 + 2 * 200000);
    int*      IDXb   = (int*)(ws + 16384 + 3 * 200000);
    int*      INVb   = (int*)(ws + 16384 + 4 * 200000);
    int*      SRCb   = (int*)(ws + 16384 + 5 * 200000);
    int*      DSTb   = (int*)(ws + 16384 + 5 * 200000 + 1 * (size_t)EE * 4);
    int*      EMb    = (int*)(ws + 16384 + 5 * 200000 + 2 * (size_t)EE * 4);
    const size_t FB = 16u * 1024u * 1024u;
    float* XA   = (float*)(ws + FB + 0 * ND * 4);
    float* XB   = (float*)(ws + FB + 1 * ND * 4);
    float* Hb   = (float*)(ws + FB + 2 * ND * 4);
    float* Ob   = (float*)(ws + FB + 3 * ND * 4);
    float* HIDb = (float*)(ws + FB + 4 * ND * 4);
    float* MXb  = (float*)(ws + FB + 5 * ND * 4);
    float* DENb = (float*)(ws + FB + 6 * ND * 4);
    float* NUMb = (float*)(ws + FB + 7 * ND * 4);

    const float* x_in  = (const float*)d_in[0];
    const float* eattr = (const float*)d_in[1];
    const int*   eidx  = (const int*)d_in[2];
    // params flattened in setup_inputs() insertion order: 21 leaves per block
    auto L = [&](int bi, int li) { return (const float*)d_in[3 + 21 * bi + li]; };

    struct GcP { const float *Wsrc, *Wedge, *t, *W1, *b1, *W2, *b2; };

    auto genconv = [&](const float* Ain, int n, int din, int dout, GcP p,
                       float* outp, const float* residual) {
        dim3 g0((n + 127) / 128, dout / 32);
        k_gemm<<<g0, 256, 0, stream>>>(Ain, p.Wsrc, nullptr, nullptr, Hb, n, din, dout, 0);
        int nd = n * dout;
        int fb = (nd + 255) / 256;
        k_fill_f<<<fb, 256, 0, stream>>>(MXb, 0.f, nd);
        k_fill_f<<<fb, 256, 0, stream>>>(DENb, 0.f, nd);
        k_fill_f<<<fb, 256, 0, stream>>>(NUMb, 0.f, nd);
        dim3 ge((EE + 255) / 256, dout / 4);
        k_edge_max<<<ge, 256, 0, stream>>>(SRCb, DSTb, EMb, eattr, Hb, p.Wedge, p.t,
                                           MXb, EE, dout);
        k_edge_sum<<<ge, 256, 0, stream>>>(SRCb, DSTb, EMb, eattr, Hb, p.Wedge, p.t,
                                           MXb, DENb, NUMb, EE, dout);
        k_node_o<<<fb, 256, 0, stream>>>(Hb, NUMb, DENb, Ob, nd);
        dim3 g1((n + 127) / 128, (2 * dout) / 32);
        k_gemm<<<g1, 256, 0, stream>>>(Ob, p.W1, p.b1, nullptr, HIDb, n, dout, 2 * dout, 1);
        dim3 g2((n + 127) / 128, dout / 32);
        k_gemm<<<g2, 256, 0, stream>>>(HIDb, p.W2, p.b2, residual, outp, n, 2 * dout, dout, 0);
    };

    // init mutable edge arrays
    k_copy_edges<<<(EE + 255) / 256, 256, 0, stream>>>(eidx, SRCb, DSTb, EMb, EE);

    const int HID[3] = {64, 96, 128};
    int n = MAXN;
    int din = 4;
    const float* cur = x_in;     // current node features (const view)
    float* bufs[2] = {XA, XB};
    int curBuf = 1;              // next output goes to XA first

    for (int bi = 0; bi < 3; ++bi) {
        const int h = HID[bi];
        GcP c0 = {L(bi, 0), L(bi, 1), L(bi, 2), L(bi, 3), L(bi, 4), L(bi, 5), L(bi, 6)};
        const float* skip_g = L(bi, 7);
        const float* skip_b = L(bi, 8);
        GcP sc = {L(bi, 9), L(bi, 10), L(bi, 11), L(bi, 12), L(bi, 13), L(bi, 14), L(bi, 15)};
        const float* regu_g = L(bi, 16);
        const float* regu_b = L(bi, 17);
        const float* pool_w = L(bi, 18);

        float* Xn = bufs[curBuf ^ 1];                    // conv0 output buffer
        // conv0 (width change)
        genconv(cur, n, din, h, c0, Xn, nullptr);
        // skip conv: hpre = relu(LN(x)); x = x + genconv(hpre)
        k_layernorm<<<(n + 7) / 8, 256, 0, stream>>>(Xn, Ob, skip_g, skip_b, n, h, 1);
        genconv(Ob, n, h, h, sc, Xn, Xn);
        // ReguBlock: relu(LN(x)) -> other buffer
        float* Xr = bufs[curBuf];
        k_layernorm<<<(n + 7) / 8, 256, 0, stream>>>(Xn, Xr, regu_g, regu_b, n, h, 1);

        // TopK pooling
        int kk = (n + 1) / 2;
        k_wnorm<<<1, 128, 0, stream>>>(pool_w, h, ctrlF);
        k_score<<<(n + 255) / 256, 256, 0, stream>>>(Xr, pool_w, ctrlF, SCOREb, KEYSb, n, h);
        k_topk_init<<<1, 256, 0, stream>>>(ctrlU, HIST);
        for (int lv = 0; lv < 4; ++lv) {
            k_hist<<<(n + 255) / 256, 256, 0, stream>>>(KEYSb, HIST, ctrlU, n, lv);
            k_hist_step<<<1, 256, 0, stream>>>(HIST, ctrlU, kk, lv);
        }
        int nb = (n + 255) / 256;
        k_sel_count<<<nb, 256, 0, stream>>>(KEYSb, ctrlU, BCG, BCT, n);
        k_sel_scan<<<1, 1, 0, stream>>>(BCG, BCT, BBG, BBT, ctrlU, nb);
        k_sel_write<<<nb, 256, 0, stream>>>(KEYSb, SCOREb, ctrlU, BBG, BBT, IDXb, VALSb, n);
        k_fill_i<<<(n + 255) / 256, 256, 0, stream>>>(INVb, -1, n);
        float* Xp = bufs[curBuf ^ 1];                    // pooled features
        k_gather_gate<<<kk, 64, 0, stream>>>(Xr, Xp, IDXb, VALSb, INVb, h);
        k_relabel<<<(EE + 255) / 256, 256, 0, stream>>>(SRCb, DSTb, EMb, INVb, EE);

        cur = Xp;
        n = kk;
        din = h;
        // next block writes into the buffer not holding `cur`
        curBuf ^= 1;  // cur now lives in bufs[curBuf^1]; keep alternation consistent
    }

    // head: mean pool (n x 128) -> 128->96 relu -> 96->64 relu -> 64->2
    k_head<<<1, 256, 0, stream>>>(cur, n,
                                  L(2, 19), L(2, 20),   // blocks[2].lin 128->96
                                  L(1, 19), L(1, 20),   // blocks[1].lin 96->64
                                  L(0, 19), L(0, 20),   // blocks[0].lin 64->2
                                  (float*)d_out);
}